// DotProductAttention_82179904242093
// MI455X (gfx1250) — compile-verified
//
#include <hip/hip_runtime.h>
#include <math.h>

typedef __attribute__((ext_vector_type(2))) float v2f;
typedef __attribute__((ext_vector_type(8))) float v8f;

#define B_    512
#define DK_   256
#define DV_   256
#define P_    2048
#define SCALE_ 0.0625f   // 1/sqrt(256)

__global__ __launch_bounds__(256) void
dpa_kernel(const float* __restrict__ Q, const float* __restrict__ K,
           const float* __restrict__ V, const int* __restrict__ mask,
           float* __restrict__ out)
{
    __shared__ float qs[DK_];   // query row
    __shared__ float es[P_];    // energy -> attn
    __shared__ float red[256];  // block reduction

    const int tid  = threadIdx.x;
    const int lane = tid & 31;
    const int wave = tid >> 5;
    const int b    = blockIdx.x;

    const int mrow = lane & 15;           // row-within-half for A; col for B
    const int hi   = (lane >= 16) ? 1 : 0;
    const int koff = hi ? 2 : 0;          // K-slot offset per A/B layout

    // Load Q[b,:] into LDS
    qs[tid] = Q[(size_t)b * DK_ + tid];
    __syncthreads();

    // ---------------- Phase 1: energy[p] = sum_d Q[d] * K[d][p] -------------
    // One wave per 16-wide p tile; A = Q replicated across 16 rows (from LDS),
    // B = K[d0:d0+4, p0:p0+16] streamed from HBM.
    const float* Kb = K + (size_t)b * DK_ * P_;
    for (int t = wave; t < (P_ / 16); t += 8) {
        const int p0 = t * 16;
        v8f acc = {0.f, 0.f, 0.f, 0.f, 0.f, 0.f, 0.f, 0.f};
        const float* kcol = Kb + (size_t)koff * P_ + p0 + mrow;
        for (int d0 = 0; d0 < DK_; d0 += 4) {
            v2f a = *(const v2f*)&qs[d0 + koff];          // K=koff, koff+1
            const float* kp = kcol + (size_t)d0 * P_;
            v2f bb;
            bb.x = __builtin_nontemporal_load(kp);        // row d0+koff
            bb.y = __builtin_nontemporal_load(kp + P_);   // row d0+koff+1
            acc = __builtin_amdgcn_wmma_f32_16x16x4_f32(
                false, a, false, bb, (short)0, acc, false, false);
        }
        // All 16 output rows identical; row 0 lives in acc[0], lanes 0..15
        if (lane < 16) es[p0 + lane] = acc[0];
    }
    __syncthreads();

    // ---------------- Phase 2: masked (multiplicative) scaled softmax -------
    const int* mb = mask + (size_t)b * P_;
    float vals[8];
    float lmax = -3.402823466e38f;
#pragma unroll
    for (int i = 0; i < 8; ++i) {
        const int p = tid + i * 256;
        const float s = es[p] * (float)mb[p] * SCALE_;
        vals[i] = s;
        lmax = fmaxf(lmax, s);
    }
    red[tid] = lmax;
    __syncthreads();
    for (int off = 128; off > 0; off >>= 1) {
        if (tid < off) red[tid] = fmaxf(red[tid], red[tid + off]);
        __syncthreads();
    }
    const float gmax = red[0];
    __syncthreads();

    float lsum = 0.f;
#pragma unroll
    for (int i = 0; i < 8; ++i) {
        const int p = tid + i * 256;
        const float e = __expf(vals[i] - gmax);
        es[p] = e;
        lsum += e;
    }
    red[tid] = lsum;
    __syncthreads();
    for (int off = 128; off > 0; off >>= 1) {
        if (tid < off) red[tid] += red[tid + off];
        __syncthreads();
    }
    const float inv = 1.0f / red[0];
    __syncthreads();
#pragma unroll
    for (int i = 0; i < 8; ++i) es[tid + i * 256] *= inv;
    __syncthreads();

    // ---------------- Phase 3: out[v] = sum_p attn[p] * V[v][p] -------------
    // A = V tile (16 v-rows x 4 p) streamed from HBM (fully utilized);
    // B column 0 = attn[p0:p0+4]; result column 0 = out tile.
    const float* Vb = V + (size_t)b * DV_ * P_;
    for (int t = wave * 2; t < wave * 2 + 2; ++t) {
        const int v0 = t * 16;
        v8f acc = {0.f, 0.f, 0.f, 0.f, 0.f, 0.f, 0.f, 0.f};
        const float* vp = Vb + (size_t)(v0 + mrow) * P_ + koff;
        for (int p0 = 0; p0 < P_; p0 += 4) {
            v2f a = __builtin_nontemporal_load((const v2f*)(vp + p0));
            v2f bb;
            bb.x = (mrow == 0) ? es[p0 + koff]     : 0.f;
            bb.y = (mrow == 0) ? es[p0 + koff + 1] : 0.f;
            acc = __builtin_amdgcn_wmma_f32_16x16x4_f32(
                false, a, false, bb, (short)0, acc, false, false);
        }
        // Column 0 of D: lane 0 has rows 0..7 (acc[0..7]), lane 16 rows 8..15
        if (mrow == 0) {
            float* op = out + (size_t)b * DV_ + v0 + (hi ? 8 : 0);
#pragma unroll
            for (int r = 0; r < 8; ++r) op[r] = acc[r];
        }
    }
}

extern "C" void kernel_launch(void* const* d_in, const int* in_sizes, int n_in,
                              void* d_out, int out_size, void* d_ws, size_t ws_size,
                              hipStream_t stream)
{
    const float* Q    = (const float*)d_in[0];
    const float* K    = (const float*)d_in[1];
    const float* V    = (const float*)d_in[2];
    const int*   mask = (const int*)d_in[3];
    float*       out  = (float*)d_out;
    (void)in_sizes; (void)n_in; (void)out_size; (void)d_ws; (void)ws_size;

    dpa_kernel<<<dim3(B_), dim3(256), 0, stream>>>(Q, K, V, mask, out);
}